// PPRGo_26474178413283
// MI455X (gfx1250) — compile-verified
//
#include <hip/hip_runtime.h>
#include <hip/hip_bf16.h>

// ---------------------------------------------------------------------------
// PPRGo on MI455X (gfx1250, wave32)
//  - MLP GEMMs via V_WMMA_F32_16X16X4_F32 (full fp32, matches fp32 reference)
//  - propagation via gather + global_atomic_add_f32 (memory/L2 bound)
// ---------------------------------------------------------------------------

#define F_IN  512
#define HID   256
#define CDIM  64
#define KPROP 10
#define ALPHA 0.1f
#define BETA  0.9f

typedef float v2f __attribute__((ext_vector_type(2)));
typedef float v8f __attribute__((ext_vector_type(8)));

// ---------------------------------------------------------------------------
// GEMM 1: H1 = relu(X[N,512] @ W1[512,256] + b1)
// 8 waves/block: 2 row-tiles (16 rows each) x 4 col-tiles (64 cols each).
// Each wave owns a 16x64 tile -> 4 accumulators, K-loop in steps of 4.
// ---------------------------------------------------------------------------
__global__ void __launch_bounds__(256) mlp1_wmma(const float* __restrict__ X,
                                                 const float* __restrict__ W1,
                                                 const float* __restrict__ b1,
                                                 float* __restrict__ H1, int n) {
  const int lane = threadIdx.x & 31;
  const int wave = threadIdx.x >> 5;
  const int mtile = blockIdx.x * 2 + (wave >> 2);   // wave-uniform
  const int m0 = mtile * 16;
  if (m0 >= n) return;                              // uniform guard, EXEC stays full
  const int n0 = (wave & 3) * 64;

  const int mrow  = lane & 15;         // A: M index / B,C,D: N index within frag
  const int khalf = (lane >> 4) << 1;  // A/B: lanes 16-31 hold K+2,K+3

  v8f acc[4] = {v8f{}, v8f{}, v8f{}, v8f{}};

  const float* __restrict__ arow = X + (size_t)(m0 + mrow) * F_IN;
  for (int k = 0; k < F_IN; k += 4) {
    __builtin_prefetch(arow + k + 64, 0, 0);        // global_prefetch_b8
    v2f a;
    a.x = arow[k + khalf + 0];
    a.y = arow[k + khalf + 1];
    const float* __restrict__ bptr = W1 + (size_t)(k + khalf) * HID + n0 + mrow;
#pragma unroll
    for (int f = 0; f < 4; ++f) {
      v2f b;
      b.x = bptr[f * 16];          // row (k+khalf)  , col n0+f*16+mrow
      b.y = bptr[f * 16 + HID];    // row (k+khalf+1), same col
      acc[f] = __builtin_amdgcn_wmma_f32_16x16x4_f32(
          false, a, false, b, (short)0, acc[f], false, false);
    }
  }

  const int rbase = m0 + ((lane >> 4) << 3);  // C/D: lanes 16-31 hold rows M+8
#pragma unroll
  for (int f = 0; f < 4; ++f) {
    const int col  = n0 + f * 16 + mrow;
    const float bias = b1[col];
#pragma unroll
    for (int j = 0; j < 8; ++j) {
      float v = acc[f][j] + bias;
      H1[(size_t)(rbase + j) * HID + col] = fmaxf(v, 0.0f);
    }
  }
}

// ---------------------------------------------------------------------------
// GEMM 2: X0 = H1[N,256] @ W2[256,64] + b2
// 8 waves/block: each wave one 16-row tile, covering all 64 output cols.
// ---------------------------------------------------------------------------
__global__ void __launch_bounds__(256) mlp2_wmma(const float* __restrict__ H1,
                                                 const float* __restrict__ W2,
                                                 const float* __restrict__ b2,
                                                 float* __restrict__ X0, int n) {
  const int lane = threadIdx.x & 31;
  const int wave = threadIdx.x >> 5;
  const int mtile = blockIdx.x * 8 + wave;
  const int m0 = mtile * 16;
  if (m0 >= n) return;

  const int mrow  = lane & 15;
  const int khalf = (lane >> 4) << 1;

  v8f acc[4] = {v8f{}, v8f{}, v8f{}, v8f{}};

  const float* __restrict__ arow = H1 + (size_t)(m0 + mrow) * HID;
  for (int k = 0; k < HID; k += 4) {
    v2f a;
    a.x = arow[k + khalf + 0];
    a.y = arow[k + khalf + 1];
    const float* __restrict__ bptr = W2 + (size_t)(k + khalf) * CDIM + mrow;
#pragma unroll
    for (int f = 0; f < 4; ++f) {
      v2f b;
      b.x = bptr[f * 16];
      b.y = bptr[f * 16 + CDIM];
      acc[f] = __builtin_amdgcn_wmma_f32_16x16x4_f32(
          false, a, false, b, (short)0, acc[f], false, false);
    }
  }

  const int rbase = m0 + ((lane >> 4) << 3);
#pragma unroll
  for (int f = 0; f < 4; ++f) {
    const int col = f * 16 + mrow;
    const float bias = b2[col];
#pragma unroll
    for (int j = 0; j < 8; ++j) {
      X0[(size_t)(rbase + j) * CDIM + col] = acc[f][j] + bias;
    }
  }
}

// ---------------------------------------------------------------------------
// GCN normalization helpers (self-loops folded in: deg starts at 1)
// ---------------------------------------------------------------------------
__global__ void deg_init(float* __restrict__ deg, int n) {
  int i = blockIdx.x * blockDim.x + threadIdx.x;
  if (i < n) deg[i] = 1.0f;  // self-loop
}

__global__ void deg_scatter(const long long* __restrict__ col,
                            float* __restrict__ deg, int e) {
  int i = blockIdx.x * blockDim.x + threadIdx.x;
  if (i < e) unsafeAtomicAdd(&deg[(int)col[i]], 1.0f);
}

__global__ void dinv_kernel(float* __restrict__ deg, int n) {
  int i = blockIdx.x * blockDim.x + threadIdx.x;
  if (i < n) deg[i] = rsqrtf(deg[i]);  // deg >= 1 always
}

__global__ void wnorm_kernel(const long long* __restrict__ row,
                             const long long* __restrict__ col,
                             const float* __restrict__ dinv,
                             float* __restrict__ wn, int e) {
  int i = blockIdx.x * blockDim.x + threadIdx.x;
  if (i < e) wn[i] = BETA * dinv[(int)row[i]] * dinv[(int)col[i]];
}

// ---------------------------------------------------------------------------
// Propagation step:
//   out = ALPHA*x0 + BETA*dinv^2*zin  (self-loop term, no atomics)
//   out[col] += BETA*norm_e * zin[row]  (edge scatter, f32 atomics in L2)
// ---------------------------------------------------------------------------
__global__ void prop_init(float* __restrict__ out, const float* __restrict__ x0,
                          const float* __restrict__ zin,
                          const float* __restrict__ dinv, int total) {
  int t = blockIdx.x * blockDim.x + threadIdx.x;
  if (t >= total) return;
  float d = dinv[t >> 6];  // node index (64 channels)
  out[t] = ALPHA * x0[t] + BETA * d * d * zin[t];
}

__global__ void prop_scatter(const long long* __restrict__ row,
                             const long long* __restrict__ col,
                             const float* __restrict__ wn,
                             const float* __restrict__ zin,
                             float* __restrict__ out, int e) {
  int t = blockIdx.x * blockDim.x + threadIdx.x;
  if (t >= e * 16) return;
  const int eidx = t >> 4;
  const int c4   = (t & 15) << 2;  // 4 channels per thread
  const int r = (int)row[eidx];
  const int c = (int)col[eidx];
  const float w = wn[eidx];
  const float4 z = *(const float4*)(zin + (size_t)r * CDIM + c4);
  float* o = out + (size_t)c * CDIM + c4;
  unsafeAtomicAdd(o + 0, w * z.x);
  unsafeAtomicAdd(o + 1, w * z.y);
  unsafeAtomicAdd(o + 2, w * z.z);
  unsafeAtomicAdd(o + 3, w * z.w);
}

// ---------------------------------------------------------------------------
extern "C" void kernel_launch(void* const* d_in, const int* in_sizes, int n_in,
                              void* d_out, int out_size, void* d_ws, size_t ws_size,
                              hipStream_t stream) {
  const float* x  = (const float*)d_in[0];
  const float* W1 = (const float*)d_in[1];
  const float* b1 = (const float*)d_in[2];
  const float* W2 = (const float*)d_in[3];
  const float* b2 = (const float*)d_in[4];
  const long long* eidx = (const long long*)d_in[5];  // int64 per reference

  const int n = in_sizes[0] / F_IN;  // 50000
  const int e = in_sizes[5] / 2;     // 800000
  const long long* erow = eidx;
  const long long* ecol = eidx + e;

  // workspace layout (floats): h1 | x0 | zA | zB | dinv | wnorm
  float* h1   = (float*)d_ws;
  float* x0   = h1 + (size_t)n * HID;
  float* zA   = x0 + (size_t)n * CDIM;
  float* zB   = zA + (size_t)n * CDIM;
  float* dinv = zB + (size_t)n * CDIM;
  float* wn   = dinv + n;

  const dim3 blk(256);
  const int mtiles = (n + 15) / 16;  // 3125

  mlp1_wmma<<<(mtiles + 1) / 2, blk, 0, stream>>>(x, W1, b1, h1, n);
  mlp2_wmma<<<(mtiles + 7) / 8, blk, 0, stream>>>(h1, W2, b2, x0, n);

  deg_init   <<<(n + 255) / 256, blk, 0, stream>>>(dinv, n);
  deg_scatter<<<(e + 255) / 256, blk, 0, stream>>>(ecol, dinv, e);
  dinv_kernel<<<(n + 255) / 256, blk, 0, stream>>>(dinv, n);
  wnorm_kernel<<<(e + 255) / 256, blk, 0, stream>>>(erow, ecol, dinv, wn, e);

  const int tot = n * CDIM;
  const float* zin = x0;
  for (int k = 0; k < KPROP; ++k) {
    float* zout = (k == KPROP - 1) ? (float*)d_out : ((k & 1) ? zB : zA);
    prop_init   <<<(tot + 255) / 256,            blk, 0, stream>>>(zout, x0, zin, dinv, tot);
    prop_scatter<<<((size_t)e * 16 + 255) / 256, blk, 0, stream>>>(erow, ecol, wn, zin, zout, e);
    zin = zout;
  }
}